// RandomWindowSlicing_69217692942480
// MI455X (gfx1250) — compile-verified
//
#include <hip/hip_runtime.h>
#include <stdint.h>

// Problem constants (match reference)
#define B_DIM   128
#define C_DIM   8
#define T_DIM   32768
#define TARGETN 29492
#define CHUNK   1024                 // output samples per workgroup
#define NCHUNK  (T_DIM / CHUNK)      // 32
#define THREADS 256                  // 8 wave32
#define LDS_STRIDE 944               // floats per channel slab (mult of 4 -> 16B aligned)

__global__ __launch_bounds__(THREADS)
void RandomWindowSlicing_kernel(const float* __restrict__ x,
                                const int*   __restrict__ starts,
                                float*       __restrict__ out) {
    __shared__ float smem[C_DIM * LDS_STRIDE];   // ~30 KB of 320 KB WGP LDS

    const int tid   = threadIdx.x;
    const int chunk = blockIdx.x;
    const int b     = blockIdx.y;
    const int t0    = chunk * CHUNK;

    const float step = (float)TARGETN / (float)(T_DIM - 1);   // ~0.90005
    const int   s    = starts[b];                              // uniform (s_load)

    // Source-index window this chunk needs: [i0_min, i1_max]
    const float q_lo = fminf((float)t0 * step, (float)(TARGETN - 1));
    const float q_hi = fminf((float)(t0 + CHUNK - 1) * step, (float)(TARGETN - 1));
    const int i0_min = (int)q_lo;
    const int i0_max = (int)q_hi;
    const int i1_max = min(i0_max + 1, TARGETN - 1);

    const int g_lo  = s + i0_min;        // first absolute element needed
    const int a_lo  = g_lo & ~3;         // align down to 16 B
    const int front = g_lo - a_lo;       // 0..3
    const int span  = (s + i1_max) - a_lo + 1;   // <= ~926  (< LDS_STRIDE)
    const int nvec  = (span + 3) >> 2;   // # of 16 B async loads per channel

    // ---- Stage: global -> LDS via CDNA5 async loads (ASYNCcnt tracked) ----
    for (int c = 0; c < C_DIM; ++c) {
        const float* row = x + (size_t)(b * C_DIM + c) * T_DIM;
        for (int v = tid; v < nvec; v += THREADS) {
            int gidx = a_lo + 4 * v;
            gidx = min(gidx, T_DIM - 4);            // clamp tail pad inside the row
            const uint64_t gptr = (uint64_t)(const void*)(row + gidx);
            const uint32_t lptr = (uint32_t)(size_t)&smem[c * LDS_STRIDE + 4 * v];
            asm volatile("global_load_async_to_lds_b128 %0, %1, off"
                         :: "v"(lptr), "v"(gptr)
                         : "memory");
        }
    }
    asm volatile("s_wait_asynccnt 0" ::: "memory");  // my wave's async loads landed
    __syncthreads();                                 // everyone's landed

    // ---- Compute: lerp from LDS (weight shared across all 8 channels) ----
#pragma unroll
    for (int k = 0; k < CHUNK / THREADS; ++k) {
        const int   t  = t0 + k * THREADS + tid;
        const float q  = fminf((float)t * step, (float)(TARGETN - 1));
        const int   i0 = (int)q;                  // q >= 0, so trunc == floor
        const float w  = q - (float)i0;
        const int   i1 = min(i0 + 1, TARGETN - 1);
        const int   j0 = front + (i0 - i0_min);
        const int   j1 = front + (i1 - i0_min);
#pragma unroll
        for (int c = 0; c < C_DIM; ++c) {
            const float v0 = smem[c * LDS_STRIDE + j0];
            const float v1 = smem[c * LDS_STRIDE + j1];
            const float r  = v0 * (1.0f - w) + v1 * w;
            // streaming store: output is written once, never re-read
            __builtin_nontemporal_store(r, &out[(size_t)(b * C_DIM + c) * T_DIM + t]);
        }
    }
}

extern "C" void kernel_launch(void* const* d_in, const int* in_sizes, int n_in,
                              void* d_out, int out_size, void* d_ws, size_t ws_size,
                              hipStream_t stream) {
    (void)in_sizes; (void)n_in; (void)d_ws; (void)ws_size; (void)out_size;
    const float* x      = (const float*)d_in[0];   // (B, C, T) f32
    const int*   starts = (const int*)d_in[1];     // (B,) i32
    float*       out    = (float*)d_out;           // (B, C, T) f32

    dim3 grid(NCHUNK, B_DIM);
    dim3 block(THREADS);
    RandomWindowSlicing_kernel<<<grid, block, 0, stream>>>(x, starts, out);
}